// DecodeBox_89369679495451
// MI455X (gfx1250) — compile-verified
//
#include <hip/hip_runtime.h>

// ---------------------------------------------------------------------------
// DecodeBox (YOLO-3D box decode): pure streaming transform + transpose.
//   in : [B=16][A*ATTRS=18][48*48*48] f32   (127.4 MB)
//   out: [B=16][A*48^3=331776][6]     f32   (127.4 MB)
// Memory-bound: ~255 MB @ 23.3 TB/s -> ~11 us roofline. No matmul -> WMMA is
// inapplicable. CDNA5 path: TDM tensor_load_to_lds pulls each block's
// 6-row strided input tile (6 x 1KB, row stride PLANE*4B) into LDS with ONE
// instruction (TENSORcnt), falling back to async global->LDS b32 (ASYNCcnt);
// bank-conflict-free LDS transpose; wide B128 non-temporal stores.
// ---------------------------------------------------------------------------

typedef float v4f __attribute__((ext_vector_type(4)));
typedef unsigned int u32x4 __attribute__((ext_vector_type(4)));
typedef int i32x4 __attribute__((ext_vector_type(4)));
typedef int i32x8 __attribute__((ext_vector_type(8)));
typedef __attribute__((address_space(1))) int* gptr_t;   // global
typedef __attribute__((address_space(3))) int* lptr_t;   // LDS

#define TILE   256
#define PLANE  110592   // 48*48*48
#define NPRED  331776   // 3*PLANE

#if __has_builtin(__builtin_amdgcn_tensor_load_to_lds) && \
    __has_builtin(__builtin_amdgcn_s_wait_tensorcnt)
#define USE_TDM 1
#endif
#if __has_builtin(__builtin_amdgcn_global_load_async_to_lds_b32) && \
    __has_builtin(__builtin_amdgcn_s_wait_asynccnt)
#define USE_ASYNC_LDS 1
#endif

__device__ __forceinline__ float fast_sigmoid(float x) {
  // 1 / (1 + 2^(-x*log2e)) : v_exp_f32 + v_rcp_f32 on the trans pipe
  return __builtin_amdgcn_rcpf(1.0f + __builtin_amdgcn_exp2f(x * -1.44269504088896341f));
}

__global__ __launch_bounds__(256) void decode_box_kernel(
    const float* __restrict__ in, float* __restrict__ out) {
  const unsigned tile = blockIdx.x;   // 0..431  (432*256 = 48^3)
  const unsigned a    = blockIdx.y;   // 0..2    anchors
  const unsigned b    = blockIdx.z;   // 0..15   batch
  const unsigned t    = threadIdx.x;  // 0..255

  // Block-uniform base of this tile's first input row (channel 0).
  const float* rowbase = in + (size_t)(b * 18u + a * 6u) * PLANE + (size_t)tile * TILE;

  __shared__ __align__(16) float lout[TILE * 6];    // transposed output staging

  float p[6];
#if defined(USE_TDM)
  __shared__ __align__(16) float lin[TILE * 6];     // [channel][256] SoA tile
  if (t < 32u) {                                    // wave 0 only (wave-uniform)
    unsigned long long ga = (unsigned long long)(size_t)rowbase;      // byte VA
    unsigned lds0 = (unsigned)(size_t)(__attribute__((address_space(3))) float*)lin;
    // D# group0: count=1 | lds_addr | global_addr[56:0] | type=2
    u32x4 g0 = { 1u, lds0, (unsigned)ga,
                 (unsigned)((ga >> 32) & 0x01FFFFFFull) | 0x80000000u };
    // D# group1: data_size=4B; tensor_dim0=110592; tensor_dim1=6;
    //            tile_dim0=256; tile_dim1=6; tensor_dim0_stride=110592
    i32x8 g1 = { (int)0x00020000u,          // [17:16] data_size=2 (4B), mask=0
                 (int)0xB0000000u,          // [31:16] tensor_dim0[15:0]=0xB000
                 (int)0x00060001u,          // [15:0] dim0[31:16]=1, [31:16] dim1=6
                 (int)0x01000000u,          // [31:16] tile_dim0=256
                 6,                         // [15:0] tile_dim1=6, tile_dim2=0
                 (int)PLANE,                // tensor_dim0_stride[31:0]
                 0, 0 };
    i32x4 gz = { 0, 0, 0, 0 };              // groups 2,3: dims 2+ unused
#if __clang_major__ >= 23
    i32x8 gz8 = { 0, 0, 0, 0, 0, 0, 0, 0 };
    __builtin_amdgcn_tensor_load_to_lds(g0, g1, gz, gz, gz8, 0);
#else
    __builtin_amdgcn_tensor_load_to_lds(g0, g1, gz, gz, 0);
#endif
    __builtin_amdgcn_s_wait_tensorcnt(0);   // data in LDS before barrier signal
  }
  __syncthreads();
#pragma unroll
  for (int c = 0; c < 6; ++c) p[c] = lin[c * TILE + t];   // bank-conflict-free
#elif defined(USE_ASYNC_LDS)
  __shared__ float lin[TILE * 6];
#pragma unroll
  for (int c = 0; c < 6; ++c) {
    // Per-lane async copy: coalesced global side, conflict-free LDS side.
    __builtin_amdgcn_global_load_async_to_lds_b32(
        (gptr_t)(rowbase + (size_t)c * PLANE + t),
        (lptr_t)&lin[c * TILE + t], 0, 0);
  }
  __builtin_amdgcn_s_wait_asynccnt(0);      // each lane reads only its own data
#pragma unroll
  for (int c = 0; c < 6; ++c) p[c] = lin[c * TILE + t];
#else
#pragma unroll
  for (int c = 0; c < 6; ++c) {
    p[c] = rowbase[(size_t)c * PLANE + t];                          // coalesced b32
    __builtin_prefetch(rowbase + (size_t)c * PLANE + t + TILE, 0, 3);
  }
#endif

  const unsigned s  = tile * TILE + t;      // spatial index in 48^3
  const unsigned gx = s % 48u;
  const unsigned gy = (s / 48u) % 48u;
  const unsigned gz = s / 2304u;
  const float aw = (float)(4u << a);        // anchor widths: 4, 8, 16

  // stride_w = stride_h = stride_d = 96/48 = 2.
  // bl = exp(l) * (anchor_w/stride_w) * stride_w = exp(l) * anchor_w.
  // LDS write addr = 6t+c: injective mod 64 over 32 lanes -> conflict-free.
  lout[t * 6 + 0] = (fast_sigmoid(p[0]) + (float)gx) * 2.0f;
  lout[t * 6 + 1] = (fast_sigmoid(p[1]) + (float)gy) * 2.0f;
  lout[t * 6 + 2] = (fast_sigmoid(p[2]) + (float)gz) * 2.0f;
  lout[t * 6 + 3] = __builtin_amdgcn_exp2f(p[3] * 1.44269504088896341f) * aw;
  lout[t * 6 + 4] = fast_sigmoid(p[4]);
  lout[t * 6 + 5] = fast_sigmoid(p[5]);

  __syncthreads();

  // Block's output is one contiguous 1536-float (6 KB) region:
  // 384 float4 chunks -> coalesced B128 non-temporal stores.
  float* outp = out + ((size_t)b * NPRED + (size_t)a * PLANE + (size_t)tile * TILE) * 6u;
  const v4f* l4 = (const v4f*)lout;
  v4f* o4 = (v4f*)outp;
  __builtin_nontemporal_store(l4[t], o4 + t);
  if (t < 128u) __builtin_nontemporal_store(l4[256u + t], o4 + 256u + t);
}

extern "C" void kernel_launch(void* const* d_in, const int* in_sizes, int n_in,
                              void* d_out, int out_size, void* d_ws, size_t ws_size,
                              hipStream_t stream) {
  (void)in_sizes; (void)n_in; (void)out_size; (void)d_ws; (void)ws_size;
  const float* in = (const float*)d_in[0];
  float* out = (float*)d_out;
  dim3 grid(432, 3, 16), block(256, 1, 1);
  decode_box_kernel<<<grid, block, 0, stream>>>(in, out);
}